// StageNet_57767310131598
// MI455X (gfx1250) — compile-verified
//
#include <hip/hip_runtime.h>
#include <math.h>

#define BB 2
#define VV 2
#define CC 32
#define DD 48
#define HH 128
#define WW 160
#define HWW (HH*WW)          /* 20480 */
#define BHWW (BB*HWW)        /* 40960 */

typedef float v2f __attribute__((ext_vector_type(2)));
typedef float v8f __attribute__((ext_vector_type(8)));

static __device__ __forceinline__ int iclamp(int x, int lo, int hi) {
    return x < lo ? lo : (x > hi ? hi : x);
}

// ---------------------------------------------------------------------------
// Kernel P: per (b, v) compute rot(3x3) + trans(3) of src_proj_new @ inv(ref_proj_new)
// proj_matrices: (B, V+1, 2, 4, 4); slot 0 = extrinsic E, slot 1 = intrinsic K.
// M = [K3x3 @ E[:3,:4] ; 0 0 0 1]  (affine), so inverse is the affine inverse.
// ---------------------------------------------------------------------------
__global__ void proj_kernel(const float* __restrict__ P, float* __restrict__ rotT)
{
    int t = threadIdx.x;
    if (t >= BB * VV) return;
    int b = t / VV, v = t % VV;

    float Mr[3][4], Ms[3][4];
    for (int which = 0; which < 2; ++which) {
        int vv = (which == 0) ? 0 : (v + 1);
        float (*M)[4] = (which == 0) ? Mr : Ms;
        for (int r = 0; r < 3; ++r)
            for (int c = 0; c < 4; ++c) {
                float s = 0.f;
                for (int k = 0; k < 3; ++k) {
                    float Kv = P[(((b*(VV+1)+vv)*2 + 1)*4 + r)*4 + k];
                    float Ev = P[(((b*(VV+1)+vv)*2 + 0)*4 + k)*4 + c];
                    s = fmaf(Kv, Ev, s);
                }
                M[r][c] = s;
            }
    }
    float a00=Mr[0][0], a01=Mr[0][1], a02=Mr[0][2];
    float a10=Mr[1][0], a11=Mr[1][1], a12=Mr[1][2];
    float a20=Mr[2][0], a21=Mr[2][1], a22=Mr[2][2];
    float det = a00*(a11*a22-a12*a21) - a01*(a10*a22-a12*a20) + a02*(a10*a21-a11*a20);
    float id = 1.0f / det;
    float Ai[3][3];
    Ai[0][0]=(a11*a22-a12*a21)*id; Ai[0][1]=(a02*a21-a01*a22)*id; Ai[0][2]=(a01*a12-a02*a11)*id;
    Ai[1][0]=(a12*a20-a10*a22)*id; Ai[1][1]=(a00*a22-a02*a20)*id; Ai[1][2]=(a02*a10-a00*a12)*id;
    Ai[2][0]=(a10*a21-a11*a20)*id; Ai[2][1]=(a01*a20-a00*a21)*id; Ai[2][2]=(a00*a11-a01*a10)*id;
    float ti[3];
    for (int r = 0; r < 3; ++r)
        ti[r] = -(Ai[r][0]*Mr[0][3] + Ai[r][1]*Mr[1][3] + Ai[r][2]*Mr[2][3]);

    float* o = rotT + (b*VV + v) * 12;
    for (int r = 0; r < 3; ++r) {
        for (int c = 0; c < 3; ++c) {
            float s = 0.f;
            for (int k = 0; k < 3; ++k) s = fmaf(Ms[r][k], Ai[k][c], s);
            o[r*3 + c] = s;
        }
        float s = 0.f;
        for (int k = 0; k < 3; ++k) s = fmaf(Ms[r][k], ti[k], s);
        o[9 + r] = s + Ms[r][3];
    }
}

// ---------------------------------------------------------------------------
// Kernel W: one wave32 per (pixel, b, v). For each 16-depth group, each lane
// bilinearly gathers the 16 channels it feeds into the WMMA A fragment, then
// 8 chained V_WMMA_F32_16X16X4_F32 produce sim (col 0) and rdot (col 1) for
// 16 depths at once, in full f32 precision.
// ---------------------------------------------------------------------------
__global__ void warp_dot_kernel(const float* __restrict__ src_feas,
                                const float* __restrict__ ref_fea,
                                const float* __restrict__ depth_values,
                                const float* __restrict__ reg_w,
                                const float* __restrict__ rotT,
                                float* __restrict__ simW,
                                float* __restrict__ rdotW)
{
    const int lane = threadIdx.x & 31;
    const int wid  = threadIdx.x >> 5;
    const int pix  = blockIdx.x * 8 + wid;
    const int b    = blockIdx.y;
    const int v    = blockIdx.z;
    const int y    = pix / WW;
    const int x    = pix % WW;
    const int n    = lane & 15;
    const bool hi  = lane >= 16;
    const int cofs = hi ? 2 : 0;

    const float* rt = rotT + (b*VV + v) * 12;
    const float r00=rt[0], r01=rt[1], r02=rt[2];
    const float r10=rt[3], r11=rt[4], r12=rt[5];
    const float r20=rt[6], r21=rt[7], r22=rt[8];
    const float t0 =rt[9], t1 =rt[10], t2 =rt[11];
    const float fx = (float)x, fy = (float)y;
    const float rx = r00*fx + r01*fy + r02;
    const float ry = r10*fx + r11*fy + r12;
    const float rz = r20*fx + r21*fy + r22;

    const float* sb = src_feas + (size_t)((v*BB + b)*CC) * HWW;

    // B fragments (K=4 x N=16): column 0 = ref channels, column 1 = ref*reg_w
    v2f bf[8];
#pragma unroll
    for (int j = 0; j < 8; ++j) {
        int kk = 4*j + cofs;
        float rf0 = ref_fea[(size_t)(b*CC + kk    )*HWW + pix];
        float rf1 = ref_fea[(size_t)(b*CC + kk + 1)*HWW + pix];
        float g0 = reg_w[kk], g1 = reg_w[kk + 1];
        float m  = (n <= 1) ? 1.0f : 0.0f;
        float s0 = (n == 1) ? g0 : 1.0f;
        float s1 = (n == 1) ? g1 : 1.0f;
        bf[j].x = rf0 * s0 * m;
        bf[j].y = rf1 * s1 * m;
    }

    for (int g = 0; g < 3; ++g) {
        const int d = g*16 + n;
        const float depthv = depth_values[(size_t)(b*DD + d)*HWW + pix];
        const float pz = fmaf(rz, depthv, t2);
        const float z  = (fabsf(pz) < 1e-6f) ? 1e-6f : pz;
        const float px = fmaf(rx, depthv, t0) / z;
        const float py = fmaf(ry, depthv, t1) / z;
        float x0f = fminf(fmaxf(floorf(px), -1.0e8f), 1.0e8f);
        float y0f = fminf(fmaxf(floorf(py), -1.0e8f), 1.0e8f);
        const float wx = px - x0f, wy = py - y0f;
        const int x0i = (int)x0f, y0i = (int)y0f;

        int   offs[4];
        float wgt[4];
        {
            const bool vx0 = (x0i   >= 0) & (x0i   < WW);
            const bool vx1 = (x0i+1 >= 0) & (x0i+1 < WW);
            const bool vy0 = (y0i   >= 0) & (y0i   < HH);
            const bool vy1 = (y0i+1 >= 0) & (y0i+1 < HH);
            const int xc0 = iclamp(x0i,   0, WW-1), xc1 = iclamp(x0i+1, 0, WW-1);
            const int yc0 = iclamp(y0i,   0, HH-1), yc1 = iclamp(y0i+1, 0, HH-1);
            offs[0] = yc0*WW + xc0;  wgt[0] = (vx0 & vy0) ? (1.f-wx)*(1.f-wy) : 0.f;
            offs[1] = yc0*WW + xc1;  wgt[1] = (vx1 & vy0) ? wx*(1.f-wy)       : 0.f;
            offs[2] = yc1*WW + xc0;  wgt[2] = (vx0 & vy1) ? (1.f-wx)*wy       : 0.f;
            offs[3] = yc1*WW + xc1;  wgt[3] = (vx1 & vy1) ? wx*wy             : 0.f;
        }

        float wvx[8], wvy[8];
#pragma unroll
        for (int j = 0; j < 8; ++j) { wvx[j] = 0.f; wvy[j] = 0.f; }
#pragma unroll
        for (int k = 0; k < 4; ++k) {
            const float* cp = sb + offs[k];
            const float wk = wgt[k];
#pragma unroll
            for (int j = 0; j < 8; ++j) {
                const int c = 4*j + cofs;
                wvx[j] = fmaf(wk, cp[(size_t)c      * HWW], wvx[j]);
                wvy[j] = fmaf(wk, cp[(size_t)(c + 1)* HWW], wvy[j]);
            }
        }

        v8f acc = {};
#pragma unroll
        for (int j = 0; j < 8; ++j) {
            v2f a; a.x = wvx[j]; a.y = wvy[j];
            acc = __builtin_amdgcn_wmma_f32_16x16x4_f32(
                false, a, false, bf[j], (short)0, acc, false, false);
        }

        if (n <= 1) {
            float* dst = (n == 0) ? simW : rdotW;
            const size_t base = ((size_t)(v*BB + b)*DD + g*16 + (hi ? 8 : 0)) * HWW + pix;
            dst[base + (size_t)0*HWW] = acc[0];
            dst[base + (size_t)1*HWW] = acc[1];
            dst[base + (size_t)2*HWW] = acc[2];
            dst[base + (size_t)3*HWW] = acc[3];
            dst[base + (size_t)4*HWW] = acc[4];
            dst[base + (size_t)5*HWW] = acc[5];
            dst[base + (size_t)6*HWW] = acc[6];
            dst[base + (size_t)7*HWW] = acc[7];
        }
    }
}

// ---------------------------------------------------------------------------
// Kernel E: depth-softmax entropy per (v,b,pixel)
// ---------------------------------------------------------------------------
__global__ void entropy_kernel(const float* __restrict__ simW, float* __restrict__ entW)
{
    int t = blockIdx.x * blockDim.x + threadIdx.x;
    if (t >= VV*BB*HWW) return;
    int pix = t % HWW;
    int vb  = t / HWW;
    const float* s = simW + (size_t)vb * DD * HWW + pix;
    float sv[DD];
    float m = -INFINITY;
#pragma unroll
    for (int d = 0; d < DD; ++d) { sv[d] = s[(size_t)d*HWW]; m = fmaxf(m, sv[d]); }
    float se = 0.f, sd = 0.f;
#pragma unroll
    for (int d = 0; d < DD; ++d) {
        float e = expf(sv[d] - m);
        se += e;
        sd = fmaf(e, sv[d], sd);
    }
    entW[t] = (m + logf(se)) - sd / se;   // lse - sum(p * s)
}

// ---------------------------------------------------------------------------
// Kernel C: 3x3 conv(2->32) + BN + relu + 1x1 conv(32->1) + sigmoid -> vw
// ---------------------------------------------------------------------------
__global__ void vis_kernel(const float* __restrict__ entW,
                           const float* __restrict__ ref_nc,
                           const float* __restrict__ w1,
                           const float* __restrict__ gamma,
                           const float* __restrict__ beta,
                           const float* __restrict__ w2,
                           const float* __restrict__ b2,
                           float* __restrict__ vwW)
{
    int t = blockIdx.x * blockDim.x + threadIdx.x;
    if (t >= VV*BB*HWW) return;
    int pix = t % HWW;
    int vb  = t / HWW;
    int b   = vb % BB;
    int y = pix / WW, x = pix % WW;
    const float* E = entW + (size_t)vb * HWW;
    const float* N = ref_nc + (size_t)b * HWW;

    float ein[9], nin[9];
    int k = 0;
#pragma unroll
    for (int dy = -1; dy <= 1; ++dy)
#pragma unroll
        for (int dx = -1; dx <= 1; ++dx, ++k) {
            int yy = y + dy, xx = x + dx;
            bool in = (yy >= 0) & (yy < HH) & (xx >= 0) & (xx < WW);
            int o = iclamp(yy,0,HH-1)*WW + iclamp(xx,0,WW-1);
            ein[k] = in ? E[o] : 0.f;
            nin[k] = in ? N[o] : 0.f;
        }

    const float invs = rsqrtf(1.0f + 1e-5f);
    float s = 0.f;
    for (int oc = 0; oc < 32; ++oc) {
        const float* wp = w1 + oc * 18;   // (oc, ic, 3, 3)
        float a = 0.f;
#pragma unroll
        for (int q = 0; q < 9; ++q) a = fmaf(wp[q],     ein[q], a);
#pragma unroll
        for (int q = 0; q < 9; ++q) a = fmaf(wp[9 + q], nin[q], a);
        float h = fmaf(a, gamma[oc] * invs, beta[oc]);
        h = fmaxf(h, 0.f);
        s = fmaf(w2[oc], h, s);
    }
    vwW[t] = 1.0f / (1.0f + expf(-(s + b2[0])));
}

// ---------------------------------------------------------------------------
// Kernel F: per (b,pixel) regression softmax -> depth, 4-window confidence, nc_mean
// ---------------------------------------------------------------------------
__global__ void final_kernel(const float* __restrict__ rdotW,
                             const float* __restrict__ vwW,
                             const float* __restrict__ depth_values,
                             const float* __restrict__ reg_b,
                             const float* __restrict__ ref_nc_sum,
                             const float* __restrict__ src_nc_sums,
                             float* __restrict__ out)
{
    int t = blockIdx.x * blockDim.x + threadIdx.x;
    if (t >= BHWW) return;
    int b = t / HWW, pix = t % HWW;

    float vw0 = vwW[(size_t)(0*BB + b)*HWW + pix];
    float vw1 = vwW[(size_t)(1*BB + b)*HWW + pix];
    float iden = 1.0f / (vw0 + vw1);
    const float* rd0 = rdotW + (size_t)(0*BB + b)*DD*HWW + pix;
    const float* rd1 = rdotW + (size_t)(1*BB + b)*DD*HWW + pix;
    const float rb = reg_b[0];

    float pp[DD];
    float m = -INFINITY;
#pragma unroll
    for (int d = 0; d < DD; ++d) {
        float num = fmaf(vw0, rd0[(size_t)d*HWW], vw1 * rd1[(size_t)d*HWW]);
        pp[d] = num * iden + rb;
        m = fmaxf(m, pp[d]);
    }
    float se = 0.f;
#pragma unroll
    for (int d = 0; d < DD; ++d) { pp[d] = expf(pp[d] - m); se += pp[d]; }
    float inv = 1.0f / se;
    const float* dvp = depth_values + (size_t)b*DD*HWW + pix;
    float depth = 0.f, didx = 0.f;
#pragma unroll
    for (int d = 0; d < DD; ++d) {
        float p = pp[d] * inv;
        pp[d] = p;
        depth = fmaf(p, dvp[(size_t)d*HWW], depth);
        didx  = fmaf(p, (float)d, didx);
    }
    int idx = iclamp((int)rintf(didx), 0, DD-1);
    float conf = 0.f;
    for (int j = idx - 1; j <= idx + 2; ++j)
        if (j >= 0 && j < DD) conf += pp[j];

    // nc_mean = sum_v (ref_nc_sum + src_nc_sums[v]) / 2 / V
    float ncs = 0.f;
    for (int v = 0; v < VV; ++v)
        ncs += ref_nc_sum[(size_t)b*HWW + pix] + src_nc_sums[(size_t)(v*BB + b)*HWW + pix];
    float nc = ncs * 0.5f / (float)VV;

    out[t]            = depth;
    out[BHWW + t]     = conf;
    out[2*BHWW + t]   = nc;
}

// ---------------------------------------------------------------------------
extern "C" void kernel_launch(void* const* d_in, const int* in_sizes, int n_in,
                              void* d_out, int out_size, void* d_ws, size_t ws_size,
                              hipStream_t stream)
{
    const float* ref_fea      = (const float*)d_in[0];
    const float* src_feas     = (const float*)d_in[1];
    const float* ref_nc       = (const float*)d_in[2];
    const float* ref_nc_sum   = (const float*)d_in[3];
    const float* src_nc_sums  = (const float*)d_in[4];
    const float* proj         = (const float*)d_in[5];
    const float* depth_values = (const float*)d_in[6];
    const float* vis_w1       = (const float*)d_in[7];
    const float* gamma        = (const float*)d_in[8];
    const float* beta         = (const float*)d_in[9];
    const float* vis_w2       = (const float*)d_in[10];
    const float* vis_b2       = (const float*)d_in[11];
    const float* reg_w        = (const float*)d_in[12];
    const float* reg_b        = (const float*)d_in[13];
    float* out = (float*)d_out;

    // workspace layout (floats)
    float* ws    = (float*)d_ws;
    float* rotT  = ws;                                    // 48 (pad 64)
    float* simW  = ws + 64;                               // V*B*D*HW = 3,932,160
    float* rdotW = simW  + (size_t)VV*BB*DD*HWW;          // 3,932,160
    float* entW  = rdotW + (size_t)VV*BB*DD*HWW;          // V*B*HW = 81,920
    float* vwW   = entW  + (size_t)VV*BB*HWW;             // 81,920

    proj_kernel<<<1, 32, 0, stream>>>(proj, rotT);

    dim3 gW(HWW / 8, BB, VV);   // one wave per pixel, 8 waves per block
    warp_dot_kernel<<<gW, 256, 0, stream>>>(src_feas, ref_fea, depth_values,
                                            reg_w, rotT, simW, rdotW);

    int nE = VV*BB*HWW;
    entropy_kernel<<<(nE + 255) / 256, 256, 0, stream>>>(simW, entW);

    vis_kernel<<<(nE + 255) / 256, 256, 0, stream>>>(entW, ref_nc, vis_w1, gamma,
                                                     beta, vis_w2, vis_b2, vwW);

    final_kernel<<<(BHWW + 255) / 256, 256, 0, stream>>>(rdotW, vwW, depth_values,
                                                         reg_b, ref_nc_sum,
                                                         src_nc_sums, out);
}